// GOTSim_90675349553997
// MI455X (gfx1250) — compile-verified
//
#include <hip/hip_runtime.h>

// ---------------------------------------------------------------------------
// GOTSim for MI455X (gfx1250): GCN convs + cost matrices via f32 WMMA,
// exact Jonker-Volgenant LAP in LDS (one workgroup per matrix) with
// wave32-shuffle argmin reductions (2 barriers per Dijkstra step).
// ---------------------------------------------------------------------------

typedef __attribute__((ext_vector_type(2))) float v2f;
typedef __attribute__((ext_vector_type(8))) float v8f;

#define GN1    384
#define GN2    384
#define NTOT   768           // GN1 + GN2
#define EDGES  3072
#define BIGC   99999.0f
#define LAPT   256           // threads per LAP block (8 waves of 32)
#define NWAVE  (LAPT / 32)

static __device__ __forceinline__ float finf() { return __builtin_inff(); }

// ---------------- degree / dinv ----------------
__global__ void k_init_deg(float* deg, int n) {
  int i = blockIdx.x * blockDim.x + threadIdx.x;
  if (i < n) deg[i] = 1.0f;                    // self loop contributes 1
}
__global__ void k_accum_deg(float* deg, const int* __restrict__ dst, int e) {
  int i = blockIdx.x * blockDim.x + threadIdx.x;
  if (i < e) atomicAdd(&deg[dst[i]], 1.0f);
}
__global__ void k_dinv(float* dinv, int n) {
  int i = blockIdx.x * blockDim.x + threadIdx.x;
  if (i < n) {
    float d = dinv[i];
    dinv[i] = (d > 0.0f) ? rsqrtf(d) : 0.0f;
  }
}

// ---------------- f32 WMMA GEMM (one wave per 16x16 tile) ------------------
// BT == false : C[m][n] = scale * sum_k A[m*lda+k] * B[k*ldb+n]      (A @ B)
// BT == true  : C[m][n] = scale * sum_k A[m*lda+k] * B[n*ldb+k]      (A @ B^T)
// Requires: M,N multiples of 16, K multiple of 4. Block = 32 threads (1 wave),
// no divergence -> EXEC all ones as WMMA requires.
template <bool BT>
__global__ void __launch_bounds__(32)
k_wmma_gemm_f32(const float* __restrict__ A, const float* __restrict__ B,
                float* __restrict__ C, int M, int N, int K,
                int lda, int ldb, int ldc, float scale) {
  const int n0   = blockIdx.x * 16;
  const int m0   = blockIdx.y * 16;
  const int lane = threadIdx.x;
  const int hi   = lane >> 4;        // 0 | 1
  const int l16  = lane & 15;

  v8f acc = {};
  for (int k0 = 0; k0 < K; k0 += 4) {
    // A fragment: v_j = A[m0+l16][k0 + j + 2*hi]
    v2f a;
    a.x = A[(m0 + l16) * lda + k0 + 2 * hi + 0];
    a.y = A[(m0 + l16) * lda + k0 + 2 * hi + 1];
    // B fragment: v_j = B[k0 + j + 2*hi][n0+l16]
    v2f b;
    if (BT) {
      b.x = B[(n0 + l16) * ldb + (k0 + 2 * hi + 0)];
      b.y = B[(n0 + l16) * ldb + (k0 + 2 * hi + 1)];
    } else {
      b.x = B[(k0 + 2 * hi + 0) * ldb + (n0 + l16)];
      b.y = B[(k0 + 2 * hi + 1) * ldb + (n0 + l16)];
    }
    acc = __builtin_amdgcn_wmma_f32_16x16x4_f32(
        /*neg_a=*/false, a, /*neg_b=*/false, b,
        /*c_mod=*/(short)0, acc, /*reuse_a=*/false, /*reuse_b=*/false);
  }
#pragma unroll
  for (int j = 0; j < 8; ++j)
    C[(m0 + j + 8 * hi) * ldc + (n0 + l16)] = scale * acc[j];
}

// ---------------- GCN aggregation ----------------
// out[i][f] = b[f] + dinv[i]^2 * h[i][f]   (bias + self-loop term)
__global__ void k_gcn_init(float* __restrict__ out, const float* __restrict__ h,
                           const float* __restrict__ b, const float* __restrict__ dinv,
                           int n, int F) {
  int idx = blockIdx.x * blockDim.x + threadIdx.x;
  if (idx >= n * F) return;
  int i = idx / F, f = idx - i * F;
  float di = dinv[i];
  out[idx] = b[f] + di * di * h[idx];
}
// out[dst][f] += dinv[src]*dinv[dst]*h[src][f] over all edges
__global__ void k_gcn_scatter(float* __restrict__ out, const float* __restrict__ h,
                              const int* __restrict__ src, const int* __restrict__ dst,
                              const float* __restrict__ dinv, int e, int F) {
  int idx = blockIdx.x * blockDim.x + threadIdx.x;
  if (idx >= e * F) return;
  int ei = idx / F, f = idx - ei * F;
  int s = src[ei], d = dst[ei];
  atomicAdd(&out[d * F + f], dinv[s] * dinv[d] * h[s * F + f]);
}
__global__ void k_relu(float* __restrict__ y, const float* __restrict__ x, int cnt) {
  int i = blockIdx.x * blockDim.x + threadIdx.x;
  if (i < cnt) y[i] = fmaxf(x[i], 0.0f);
}

// ---------------- cost-matrix border blocks ----------------
// S is NTOT x NTOT; main (r<GN1 && c<GN1) already written by WMMA GEMM.
__global__ void k_fill_blocks(float* __restrict__ S,
                              const float* __restrict__ f1, const float* __restrict__ f2,
                              const float* __restrict__ dp, const float* __restrict__ ip,
                              int F) {
  int idx = blockIdx.x * blockDim.x + threadIdx.x;
  if (idx >= NTOT * NTOT) return;
  int r = idx / NTOT, c = idx - r * NTOT;
  if (r < GN1 && c < GN1) return;
  float val;
  if (r < GN1) {                       // deletion block (top-right)
    int j = c - GN1;
    if (r == j) {
      float s = 0.0f;
      for (int k = 0; k < F; ++k) s += f1[r * F + k] * dp[k];
      val = -s;
    } else val = BIGC;
  } else if (c < GN1) {                // insertion block (bottom-left)
    int i = r - GN1;
    if (i == c) {
      float s = 0.0f;
      for (int k = 0; k < F; ++k) s += f2[i * F + k] * ip[k];
      val = -s;
    } else val = BIGC;
  } else {
    val = 0.0f;                        // dummy block
  }
  S[idx] = val;
}

// ---------------- exact JV LAP, one block per matrix ----------------
__global__ void __launch_bounds__(LAPT)
k_lap(const float* __restrict__ Sall, float* __restrict__ losses) {
  const float* C = Sall + (size_t)blockIdx.x * NTOT * NTOT;
  __shared__ float u[NTOT], v[NTOT], shortest[NTOT];
  __shared__ int   path[NTOT], col4row[NTOT], row4col[NTOT], SR[NTOT], SC[NTOT];
  __shared__ float wv[NWAVE];
  __shared__ int   wi[NWAVE];
  __shared__ int   s_i, s_sink;
  __shared__ float s_min;

  const int tid  = threadIdx.x;
  const int wav  = tid >> 5;
  const int lane = tid & 31;

  for (int t = 0; t < NTOT / LAPT; ++t) {
    int j = tid + t * LAPT;
    u[j] = 0.0f; v[j] = 0.0f; col4row[j] = -1; row4col[j] = -1;
  }
  __syncthreads();

  for (int row = 0; row < NTOT; ++row) {
    for (int t = 0; t < NTOT / LAPT; ++t) {
      int j = tid + t * LAPT;
      shortest[j] = finf(); path[j] = -1; SR[j] = 0; SC[j] = 0;
    }
    if (tid == 0) { s_i = row; s_sink = -1; s_min = 0.0f; }
    __syncthreads();

    while (s_sink < 0) {
      const int   i  = s_i;
      const float mv = s_min;
      const float ui = u[i];
      if (tid == 0) SR[i] = 1;
      const float* Ci = C + (size_t)i * NTOT;
      float bv = finf(); int bj = -1;
#pragma unroll
      for (int t = 0; t < NTOT / LAPT; ++t) {
        int j = tid + t * LAPT;
        if (!SC[j]) {
          float d = mv + Ci[j] - ui - v[j];
          if (d < shortest[j]) { shortest[j] = d; path[j] = i; }
          float s = shortest[j];
          if (s < bv) { bv = s; bj = j; }
        }
      }
      // wave32 min-pair reduction (no barriers)
#pragma unroll
      for (int off = 16; off > 0; off >>= 1) {
        float ov = __shfl_down(bv, off);
        int   oj = __shfl_down(bj, off);
        if (ov < bv) { bv = ov; bj = oj; }
      }
      if (lane == 0) { wv[wav] = bv; wi[wav] = bj; }
      __syncthreads();
      if (tid == 0) {
        float mbv = wv[0]; int mbj = wi[0];
#pragma unroll
        for (int w = 1; w < NWAVE; ++w)
          if (wv[w] < mbv) { mbv = wv[w]; mbj = wi[w]; }
        s_min = mbv;
        SC[mbj] = 1;
        if (row4col[mbj] < 0) s_sink = mbj;
        else                  s_i    = row4col[mbj];
      }
      __syncthreads();
    }

    // dual updates
    const float mv = s_min;
    for (int t = 0; t < NTOT / LAPT; ++t) {
      int j = tid + t * LAPT;
      if (SR[j] && j != row) u[j] += mv - shortest[col4row[j]];
      if (SC[j])             v[j] -= mv - shortest[j];
    }
    if (tid == 0) u[row] += mv;
    __syncthreads();

    // augment along path (sequential)
    if (tid == 0) {
      int j = s_sink;
      while (true) {
        int i = path[j];
        row4col[j] = i;
        int jn = col4row[i];
        col4row[i] = j;
        j = jn;
        if (i == row) break;
      }
    }
    __syncthreads();
  }

  // loss = sum_r C[r, col4row[r]] / NTOT
  float acc = 0.0f;
  for (int t = 0; t < NTOT / LAPT; ++t) {
    int r = tid + t * LAPT;
    acc += C[(size_t)r * NTOT + col4row[r]];
  }
#pragma unroll
  for (int off = 16; off > 0; off >>= 1) acc += __shfl_down(acc, off);
  if (lane == 0) wv[wav] = acc;
  __syncthreads();
  if (tid == 0) {
    float s = 0.0f;
#pragma unroll
    for (int w = 0; w < NWAVE; ++w) s += wv[w];
    losses[blockIdx.x] = s / (float)NTOT;
  }
}

// ---------------- scalar epilogue ----------------
__global__ void k_final(const float* __restrict__ losses,
                        const float* __restrict__ score_w,
                        const float* __restrict__ score_b,
                        const float* __restrict__ avg_v,
                        float* __restrict__ out) {
  if (blockIdx.x == 0 && threadIdx.x == 0) {
    float mc[3];
#pragma unroll
    for (int k = 0; k < 3; ++k) mc[k] = 2.0f * losses[k] / (float)(GN1 + GN2);
    float logit = score_w[0] * mc[0] + score_w[1] * mc[1] + score_w[2] * mc[2] + score_b[0];
    float sc = 1.0f / (1.0f + expf(-logit));
    out[0] = sc;
    out[1] = -logf(sc) * avg_v[0];
    out[2] = mc[0]; out[3] = mc[1]; out[4] = mc[2];
  }
}

// ---------------------------------------------------------------------------
static inline int cdiv(int a, int b) { return (a + b - 1) / b; }

extern "C" void kernel_launch(void* const* d_in, const int* in_sizes, int n_in,
                              void* d_out, int out_size, void* d_ws, size_t ws_size,
                              hipStream_t stream) {
  (void)in_sizes; (void)n_in; (void)out_size; (void)ws_size;

  const int*   ei1 = (const int*)d_in[0];       // (2,E): [src | dst]
  const int*   ei2 = (const int*)d_in[1];
  const float* x1  = (const float*)d_in[2];     // 384 x 32
  const float* x2  = (const float*)d_in[3];
  const float* avg_v = (const float*)d_in[6];
  const float* W[3]  = { (const float*)d_in[7],  (const float*)d_in[11], (const float*)d_in[15] };
  const float* Bv[3] = { (const float*)d_in[8],  (const float*)d_in[12], (const float*)d_in[16] };
  const float* DP[3] = { (const float*)d_in[9],  (const float*)d_in[13], (const float*)d_in[17] };
  const float* IP[3] = { (const float*)d_in[10], (const float*)d_in[14], (const float*)d_in[18] };
  const float* score_w = (const float*)d_in[19];
  const float* score_b = (const float*)d_in[20];

  // ---- workspace layout (floats) ----
  float* ws    = (float*)d_ws;
  float* dinv1 = ws;                       // 384
  float* dinv2 = dinv1 + GN1;              // 384
  float* h     = dinv2 + GN2;              // 384*128
  float* xr1   = h   + GN1 * 128;          // 384*128 (relu'd feats graph1)
  float* xr2   = xr1 + GN1 * 128;          // 384*128
  float* f1b   = xr2 + GN2 * 128;          // feats g1: 128+64+32 cols
  float* f1[3] = { f1b, f1b + GN1 * 128, f1b + GN1 * (128 + 64) };
  float* f2b   = f1b + GN1 * (128 + 64 + 32);
  float* f2[3] = { f2b, f2b + GN2 * 128, f2b + GN2 * (128 + 64) };
  float* Sbase = f2b + GN2 * (128 + 64 + 32);        // 3 * 768*768
  float* losses = Sbase + 3 * (size_t)NTOT * NTOT;   // 3

  const int TPB = 256;
  const int FO[3] = { 128, 64, 32 };

  // ---- degrees / dinv ----
  k_init_deg <<<cdiv(GN1, TPB),   TPB, 0, stream>>>(dinv1, GN1);
  k_init_deg <<<cdiv(GN2, TPB),   TPB, 0, stream>>>(dinv2, GN2);
  k_accum_deg<<<cdiv(EDGES, TPB), TPB, 0, stream>>>(dinv1, ei1 + EDGES, EDGES);
  k_accum_deg<<<cdiv(EDGES, TPB), TPB, 0, stream>>>(dinv2, ei2 + EDGES, EDGES);
  k_dinv     <<<cdiv(GN1, TPB),   TPB, 0, stream>>>(dinv1, GN1);
  k_dinv     <<<cdiv(GN2, TPB),   TPB, 0, stream>>>(dinv2, GN2);

  // ---- GCN layers ----
  const float* cur1 = x1;
  const float* cur2 = x2;
  int Fin = 32;
  for (int l = 0; l < 3; ++l) {
    const int F = FO[l];
    dim3 gg(F / 16, GN1 / 16);
    // graph 1
    k_wmma_gemm_f32<false><<<gg, 32, 0, stream>>>(cur1, W[l], h, GN1, F, Fin, Fin, F, F, 1.0f);
    k_gcn_init   <<<cdiv(GN1 * F, TPB),   TPB, 0, stream>>>(f1[l], h, Bv[l], dinv1, GN1, F);
    k_gcn_scatter<<<cdiv(EDGES * F, TPB), TPB, 0, stream>>>(f1[l], h, ei1, ei1 + EDGES, dinv1, EDGES, F);
    // graph 2 (h reused; stream order serializes)
    k_wmma_gemm_f32<false><<<gg, 32, 0, stream>>>(cur2, W[l], h, GN2, F, Fin, Fin, F, F, 1.0f);
    k_gcn_init   <<<cdiv(GN2 * F, TPB),   TPB, 0, stream>>>(f2[l], h, Bv[l], dinv2, GN2, F);
    k_gcn_scatter<<<cdiv(EDGES * F, TPB), TPB, 0, stream>>>(f2[l], h, ei2, ei2 + EDGES, dinv2, EDGES, F);
    if (l < 2) {
      k_relu<<<cdiv(GN1 * F, TPB), TPB, 0, stream>>>(xr1, f1[l], GN1 * F);
      k_relu<<<cdiv(GN2 * F, TPB), TPB, 0, stream>>>(xr2, f2[l], GN2 * F);
      cur1 = xr1; cur2 = xr2;
    }
    Fin = F;
  }

  // ---- cost matrices: main block = -(f1 @ f2^T) via WMMA, then borders ----
  for (int l = 0; l < 3; ++l) {
    float* S = Sbase + (size_t)l * NTOT * NTOT;
    dim3 gm(GN2 / 16, GN1 / 16);
    k_wmma_gemm_f32<true><<<gm, 32, 0, stream>>>(f1[l], f2[l], S, GN1, GN2, FO[l],
                                                 FO[l], FO[l], NTOT, -1.0f);
    k_fill_blocks<<<cdiv(NTOT * NTOT, TPB), TPB, 0, stream>>>(S, f1[l], f2[l], DP[l], IP[l], FO[l]);
  }

  // ---- three exact LAP solves, one workgroup each (3 WGPs in parallel) ----
  k_lap<<<3, LAPT, 0, stream>>>(Sbase, losses);

  // ---- epilogue ----
  k_final<<<1, 64, 0, stream>>>(losses, score_w, score_b, avg_v, (float*)d_out);
}